// BlockModel_30081950941763
// MI455X (gfx1250) — compile-verified
//
#include <hip/hip_runtime.h>
#include <hip/hip_bf16.h>

typedef float v2f __attribute__((ext_vector_type(2)));
typedef float v8f __attribute__((ext_vector_type(8)));

#define HID 64

// ---------------------------------------------------------------------------
// h = x @ Wn + bn   (x: [rows,2], Wn: [2,64], bn: [64])   (also used for edges)
// ---------------------------------------------------------------------------
__global__ __launch_bounds__(256)
void gcn_encode_kernel(const float* __restrict__ xin, const float* __restrict__ W,
                       const float* __restrict__ b, float* __restrict__ out, long total)
{
    long t = (long)blockIdx.x * 256 + threadIdx.x;
    if (t >= total) return;
    long r = t >> 6;
    int  k = (int)(t & 63);
    out[t] = fmaf(xin[r * 2 + 0], W[k], fmaf(xin[r * 2 + 1], W[64 + k], b[k]));
}

// ---------------------------------------------------------------------------
// zero a float range (float4 granularity)
// ---------------------------------------------------------------------------
__global__ __launch_bounds__(256)
void gcn_zero_kernel(float* __restrict__ p, long total4)
{
    long t = (long)blockIdx.x * 256 + threadIdx.x;
    if (t >= total4) return;
    float4 z = make_float4(0.f, 0.f, 0.f, 0.f);
    *(float4*)(p + t * 4) = z;
}

// ---------------------------------------------------------------------------
// Node-side transforms: hA = h@A, hB = h@B, hV = h@V, hU = h@U
// One 16-row tile per block; 4 waves, wave w owns one weight matrix.
// GEMM via V_WMMA_F32_16X16X4_F32 (wave32), K=64 in 16 steps of 4.
// ---------------------------------------------------------------------------
__global__ __launch_bounds__(128)
void gcn_node_transform_kernel(const float* __restrict__ h,
                               const float* __restrict__ Wa, const float* __restrict__ Wb,
                               const float* __restrict__ Wv, const float* __restrict__ Wu,
                               float* __restrict__ hA, float* __restrict__ hB,
                               float* __restrict__ hV, float* __restrict__ hU, int N)
{
    __shared__ float htile[16][HID];
    const int tid  = threadIdx.x;
    const int row0 = blockIdx.x * 16;

    // cooperative stage of the 16x64 h tile (1024 floats, 128 thr x 2 float4)
    #pragma unroll
    for (int t = 0; t < 2; ++t) {
        int idx = (t * 128 + tid) * 4;
        int r = idx >> 6, c = idx & 63;
        int row = row0 + r; if (row >= N) row = N - 1;
        *(float4*)(&htile[r][c]) = *(const float4*)(h + (size_t)row * HID + c);
    }
    __syncthreads();

    const int wv   = tid >> 5;     // wave id -> weight matrix
    const int lane = tid & 31;
    const int m    = lane & 15;    // A row / B col within tile
    const int kh   = lane >> 4;    // which K pair

    const float* W = (wv == 0) ? Wa : (wv == 1) ? Wb : (wv == 2) ? Wv : Wu;
    float*       O = (wv == 0) ? hA : (wv == 1) ? hB : (wv == 2) ? hV : hU;

    v8f acc0 = {}, acc1 = {}, acc2 = {}, acc3 = {};
    #pragma unroll
    for (int ks = 0; ks < 16; ++ks) {
        const int k = ks * 4 + kh * 2;
        v2f a; a.x = htile[m][k]; a.y = htile[m][k + 1];
        const float* wr0 = W + (size_t)k * HID;
        const float* wr1 = W + (size_t)(k + 1) * HID;
        v2f b0, b1, b2, b3;
        b0.x = wr0[m +  0]; b0.y = wr1[m +  0];
        b1.x = wr0[m + 16]; b1.y = wr1[m + 16];
        b2.x = wr0[m + 32]; b2.y = wr1[m + 32];
        b3.x = wr0[m + 48]; b3.y = wr1[m + 48];
        acc0 = __builtin_amdgcn_wmma_f32_16x16x4_f32(false, a, false, b0, (short)0, acc0, false, false);
        acc1 = __builtin_amdgcn_wmma_f32_16x16x4_f32(false, a, false, b1, (short)0, acc1, false, false);
        acc2 = __builtin_amdgcn_wmma_f32_16x16x4_f32(false, a, false, b2, (short)0, acc2, false, false);
        acc3 = __builtin_amdgcn_wmma_f32_16x16x4_f32(false, a, false, b3, (short)0, acc3, false, false);
    }

    // D layout: VGPR r holds M = r + 8*kh, N = nt*16 + m
    #pragma unroll
    for (int r = 0; r < 8; ++r) {
        int row = row0 + r + 8 * kh;
        if (row < N) {
            float* orow = O + (size_t)row * HID + m;
            orow[0]  = acc0[r];
            orow[16] = acc1[r];
            orow[32] = acc2[r];
            orow[48] = acc3[r];
        }
    }
}

// ---------------------------------------------------------------------------
// Edge kernel: e_hat = e@C + hA[src] + hB[dst] + cb ; sigma = sigmoid(e_hat)
//   e_out = relu(e_hat) ; atomic num[dst] += sigma*hV[src] ; den[dst] += sigma
// 4 waves/block, each wave owns one 16-edge tile. e@C via WMMA f32 16x16x4.
// ---------------------------------------------------------------------------
__global__ __launch_bounds__(128)
void gcn_edge_kernel(const float* __restrict__ ein, const float* __restrict__ Wc,
                     const float* __restrict__ cb,
                     const float* __restrict__ hA, const float* __restrict__ hB,
                     const float* __restrict__ hV,
                     const int* __restrict__ src, const int* __restrict__ dst,
                     float* __restrict__ eout, float* __restrict__ num,
                     float* __restrict__ den, int E)
{
    __shared__ float etile[4][16][HID];
    __shared__ float ehat [4][16][HID];
    __shared__ int   s_src[4][16];
    __shared__ int   s_dst[4][16];

    const int tid  = threadIdx.x;
    const int wv   = tid >> 5;
    const int lane = tid & 31;
    const int eb   = (blockIdx.x * 4 + wv) * 16;   // first edge of this wave's tile

    // stage this wave's 16x64 edge tile (32 lanes x 8 float4)
    #pragma unroll
    for (int t = 0; t < 8; ++t) {
        int idx = (t * 32 + lane) * 4;
        int r = idx >> 6, c = idx & 63;
        int ei = eb + r; if (ei >= E) ei = E - 1;
        *(float4*)(&etile[wv][r][c]) = *(const float4*)(ein + (size_t)ei * HID + c);
    }
    if (lane < 16) {
        int ei = eb + lane; if (ei >= E) ei = E - 1;
        s_src[wv][lane] = src[ei];
        s_dst[wv][lane] = dst[ei];
    }
    // prefetch next block's edge tile into cache (global_prefetch_b8)
    if (eb + 64 < E) __builtin_prefetch(ein + ((size_t)eb + 64) * HID, 0, 1);
    __syncthreads();

    const int m  = lane & 15;
    const int kh = lane >> 4;

    v8f acc0 = {}, acc1 = {}, acc2 = {}, acc3 = {};
    #pragma unroll
    for (int ks = 0; ks < 16; ++ks) {
        const int k = ks * 4 + kh * 2;
        v2f a; a.x = etile[wv][m][k]; a.y = etile[wv][m][k + 1];
        const float* wr0 = Wc + (size_t)k * HID;
        const float* wr1 = Wc + (size_t)(k + 1) * HID;
        v2f b0, b1, b2, b3;
        b0.x = wr0[m +  0]; b0.y = wr1[m +  0];
        b1.x = wr0[m + 16]; b1.y = wr1[m + 16];
        b2.x = wr0[m + 32]; b2.y = wr1[m + 32];
        b3.x = wr0[m + 48]; b3.y = wr1[m + 48];
        acc0 = __builtin_amdgcn_wmma_f32_16x16x4_f32(false, a, false, b0, (short)0, acc0, false, false);
        acc1 = __builtin_amdgcn_wmma_f32_16x16x4_f32(false, a, false, b1, (short)0, acc1, false, false);
        acc2 = __builtin_amdgcn_wmma_f32_16x16x4_f32(false, a, false, b2, (short)0, acc2, false, false);
        acc3 = __builtin_amdgcn_wmma_f32_16x16x4_f32(false, a, false, b3, (short)0, acc3, false, false);
    }

    // spill e@C to LDS, re-layout WMMA fragments -> row-major tile
    #pragma unroll
    for (int r = 0; r < 8; ++r) {
        const int M = r + 8 * kh;
        ehat[wv][M][m +  0] = acc0[r];
        ehat[wv][M][m + 16] = acc1[r];
        ehat[wv][M][m + 32] = acc2[r];
        ehat[wv][M][m + 48] = acc3[r];
    }
    __syncthreads();

    // fused gate + scatter: each lane owns 2 features for all 16 edges
    const int k0 = lane * 2;
    const float cb0 = cb[k0], cb1 = cb[k0 + 1];
    for (int el = 0; el < 16; ++el) {
        const int ei = eb + el;
        if (ei >= E) break;
        const int s = s_src[wv][el];
        const int d = s_dst[wv][el];
        const size_t so = (size_t)s * HID + k0;
        const size_t dof = (size_t)d * HID + k0;

        float v0 = ehat[wv][el][k0]     + hA[so]     + hB[dof]     + cb0;
        float v1 = ehat[wv][el][k0 + 1] + hA[so + 1] + hB[dof + 1] + cb1;
        float sg0 = 1.0f / (1.0f + __expf(-v0));
        float sg1 = 1.0f / (1.0f + __expf(-v1));

        float2 eo; eo.x = fmaxf(v0, 0.f); eo.y = fmaxf(v1, 0.f);
        *(float2*)(eout + (size_t)ei * HID + k0) = eo;   // coalesced 256B/wave row

        atomicAdd(&num[dof],     sg0 * hV[so]);
        atomicAdd(&num[dof + 1], sg1 * hV[so + 1]);
        atomicAdd(&den[dof],     sg0);
        atomicAdd(&den[dof + 1], sg1);
    }
}

// ---------------------------------------------------------------------------
// h' = relu(hU + num/(den + 1e-6))      (elementwise, float4)
// ---------------------------------------------------------------------------
__global__ __launch_bounds__(256)
void gcn_update_kernel(const float* __restrict__ hU, const float* __restrict__ num,
                       const float* __restrict__ den, float* __restrict__ hout, long total4)
{
    long t = (long)blockIdx.x * 256 + threadIdx.x;
    if (t >= total4) return;
    const long o = t * 4;
    float4 u  = *(const float4*)(hU + o);
    float4 nm = *(const float4*)(num + o);
    float4 dn = *(const float4*)(den + o);
    float4 r;
    r.x = fmaxf(u.x + nm.x / (dn.x + 1e-6f), 0.f);
    r.y = fmaxf(u.y + nm.y / (dn.y + 1e-6f), 0.f);
    r.z = fmaxf(u.z + nm.z / (dn.z + 1e-6f), 0.f);
    r.w = fmaxf(u.w + nm.w / (dn.w + 1e-6f), 0.f);
    *(float4*)(hout + o) = r;
}

// ---------------------------------------------------------------------------
// scores = concat(h[src_s], h[dst_s]) @ Wp + bp
// ---------------------------------------------------------------------------
__global__ __launch_bounds__(256)
void gcn_score_kernel(const float* __restrict__ h, const int* __restrict__ ss,
                      const int* __restrict__ ds, const float* __restrict__ Wp,
                      const float* __restrict__ bp, float* __restrict__ out, int ES)
{
    int t = blockIdx.x * 256 + threadIdx.x;
    if (t >= ES) return;
    const size_t so = (size_t)ss[t] * HID;
    const size_t dof = (size_t)ds[t] * HID;
    float acc = bp[0];
    #pragma unroll
    for (int k = 0; k < HID; k += 4) {
        float4 hs = *(const float4*)(h + so + k);
        float4 w1 = *(const float4*)(Wp + k);
        float4 hd = *(const float4*)(h + dof + k);
        float4 w2 = *(const float4*)(Wp + HID + k);
        acc += hs.x * w1.x + hs.y * w1.y + hs.z * w1.z + hs.w * w1.w;
        acc += hd.x * w2.x + hd.y * w2.y + hd.z * w2.z + hd.w * w2.w;
    }
    out[t] = acc;
}

// ---------------------------------------------------------------------------
extern "C" void kernel_launch(void* const* d_in, const int* in_sizes, int n_in,
                              void* d_out, int out_size, void* d_ws, size_t ws_size,
                              hipStream_t stream)
{
    const float* x    = (const float*)d_in[0];
    const float* ein  = (const float*)d_in[1];
    const float* Wn   = (const float*)d_in[2];
    const float* bn   = (const float*)d_in[3];
    const float* We   = (const float*)d_in[4];
    const float* be   = (const float*)d_in[5];
    const float* A    = (const float*)d_in[6];
    const float* B    = (const float*)d_in[7];
    const float* C    = (const float*)d_in[8];
    const float* cb   = (const float*)d_in[9];
    const float* U    = (const float*)d_in[10];
    const float* V    = (const float*)d_in[11];
    const float* Wp   = (const float*)d_in[12];
    const float* bp   = (const float*)d_in[13];
    const int* src    = (const int*)d_in[14];
    const int* dst    = (const int*)d_in[15];
    const int* src_s  = (const int*)d_in[16];
    const int* dst_s  = (const int*)d_in[17];

    const int N  = in_sizes[0] / 2;
    const int E  = in_sizes[1] / 2;
    const int ES = in_sizes[16];
    const int L  = in_sizes[6] / (HID * HID);

    // workspace layout
    float* ws = (float*)d_ws;
    size_t o = 0;
    const size_t nh = (size_t)N * HID;
    const size_t eh = (size_t)E * HID;
    float* h0  = ws + o; o += nh;
    float* h1  = ws + o; o += nh;
    float* hA  = ws + o; o += nh;
    float* hB  = ws + o; o += nh;
    float* hV  = ws + o; o += nh;
    float* hU  = ws + o; o += nh;
    float* num = ws + o; o += nh;
    float* den = ws + o; o += nh;   // contiguous after num: zero both at once
    float* e0  = ws + o; o += eh;
    float* e1  = ws + o; o += eh;

    // encoders
    {
        long tot = (long)N * HID;
        gcn_encode_kernel<<<(unsigned)((tot + 255) / 256), 256, 0, stream>>>(x, Wn, bn, h0, tot);
    }
    {
        long tot = (long)E * HID;
        gcn_encode_kernel<<<(unsigned)((tot + 255) / 256), 256, 0, stream>>>(ein, We, be, e0, tot);
    }

    float* hcur = h0; float* hnext = h1;
    float* ecur = e0; float* enext = e1;

    for (int i = 0; i < L; ++i) {
        // zero num & den (contiguous 2*N*H floats)
        {
            long tot4 = (long)(2 * nh) / 4;
            gcn_zero_kernel<<<(unsigned)((tot4 + 255) / 256), 256, 0, stream>>>(num, tot4);
        }
        // node-side GEMMs (gather commutes with linear map)
        gcn_node_transform_kernel<<<(unsigned)((N + 15) / 16), 128, 0, stream>>>(
            hcur, A + (size_t)i * HID * HID, B + (size_t)i * HID * HID,
            V + (size_t)i * HID * HID, U + (size_t)i * HID * HID,
            hA, hB, hV, hU, N);
        // edge GEMM + gate + scatter
        gcn_edge_kernel<<<(unsigned)((E + 63) / 64), 128, 0, stream>>>(
            ecur, C + (size_t)i * HID * HID, cb + (size_t)i * HID,
            hA, hB, hV, src, dst, enext, num, den, E);
        // node update
        {
            long tot4 = (long)nh / 4;
            gcn_update_kernel<<<(unsigned)((tot4 + 255) / 256), 256, 0, stream>>>(
                hU, num, den, hnext, tot4);
        }
        float* tp = hcur; hcur = hnext; hnext = tp;
        tp = ecur; ecur = enext; enext = tp;
    }

    // final edge scoring
    gcn_score_kernel<<<(unsigned)((ES + 255) / 256), 256, 0, stream>>>(
        hcur, src_s, dst_s, Wp, bp, (float*)d_out, ES);
}